// GNS_homogeneous_68719476736495
// MI455X (gfx1250) — compile-verified
//
#include <hip/hip_runtime.h>
#include <math.h>

// ---------------------------------------------------------------------------
// Graph transformer forward (f32) for MI455X / gfx1250.
// Large dense GEMMs use V_WMMA_F32_16X16X4_F32 (full f32, wave32). Each wave
// computes a 32x64 output block (2 M-tiles x 4 N-tiles, 8 v8f accumulators);
// K is a compile-time template parameter (64 or 256) so the K-loop has no
// remainder path and no accumulator phi-copies.
// Odd-shaped tiny GEMMs (K=6, K=2, Nout=6) use a scalar kernel.
// Edge softmax / scatter phases are bandwidth-bound: coalesced loads + native
// f32 global atomics.
// ---------------------------------------------------------------------------

constexpr int kN     = 20000;   // nodes
constexpr int kE     = 320000;  // edges
constexpr int kHeads = 4;
constexpr int kHid   = 64;
constexpr int kHC    = 256;     // kHeads * kHid

typedef __attribute__((ext_vector_type(2))) float v2f;
typedef __attribute__((ext_vector_type(8))) float v8f;

__device__ __forceinline__ float wsum32(float v) {
#pragma unroll
  for (int o = 16; o > 0; o >>= 1) v += __shfl_xor(v, o, 32);
  return v;
}

__device__ __forceinline__ float lrelu(float v) { return v > 0.f ? v : 0.01f * v; }

__device__ __forceinline__ void atomicMaxF(float* addr, float val) {
  int* ia = (int*)addr;
  int old = *ia;
  while (val > __int_as_float(old)) {
    int assumed = old;
    old = atomicCAS(ia, assumed, __float_as_int(val));
    if (old == assumed) break;
  }
}

// ---------------------------------------------------------------------------
// Fast f32 WMMA GEMM:  C[M x Nout] = act(A[M x K] @ W[K x Nout] + bias)
// REQUIRES: M % 32 == 0, Nout % 64 == 0, K in {64, 256} (template).
// Wave computes 32x64: 2 M-tiles x 4 N-tiles; per K-step 2 A b64-loads and
// 8 B b32-loads feed 8 WMMAs (each B fragment reused twice).
// A-frag (16x4 f32): lane l holds A[l%16][kb + 2*(l/16) + {0,1}]  (contiguous)
// B-frag (4x16 f32): lane l holds W[kb + 2*(l/16) + {0,1}][col]
// C/D: VGPR r -> M = r (lanes 0-15) / r+8 (lanes 16-31).
// ---------------------------------------------------------------------------
template <int K>
__global__ void gemm_f32_wmma(const float* __restrict__ A, const float* __restrict__ W,
                              const float* __restrict__ bias, float* __restrict__ C,
                              int Nout, int leaky) {
  const int lane = threadIdx.x;
  const int hi   = lane >> 4;    // half-wave 0/1
  const int lo   = lane & 15;

  const float* Ap0 = A + (size_t)(blockIdx.x * 32 + lo) * K + 2 * hi;
  const float* Ap1 = Ap0 + (size_t)16 * K;
  const float* Bp[4];
  int col[4];
#pragma unroll
  for (int t = 0; t < 4; ++t) {
    col[t] = (blockIdx.y * 4 + t) * 16 + lo;
    Bp[t]  = W + (size_t)(2 * hi) * Nout + col[t];
  }

  v8f acc[2][4];
#pragma unroll
  for (int m = 0; m < 2; ++m)
#pragma unroll
    for (int t = 0; t < 4; ++t) acc[m][t] = (v8f){0.f,0.f,0.f,0.f,0.f,0.f,0.f,0.f};

#pragma unroll 2
  for (int kb = 0; kb < K; kb += 4) {
    v2f a0 = *(const v2f*)(Ap0 + kb);
    v2f a1 = *(const v2f*)(Ap1 + kb);
    const size_t o0 = (size_t)kb * Nout;
    v2f b[4];
#pragma unroll
    for (int t = 0; t < 4; ++t) {
      b[t].x = Bp[t][o0];
      b[t].y = Bp[t][o0 + Nout];
    }
#pragma unroll
    for (int t = 0; t < 4; ++t) {
      acc[0][t] = __builtin_amdgcn_wmma_f32_16x16x4_f32(false, a0, false, b[t], (short)0,
                                                        acc[0][t], false, false);
      acc[1][t] = __builtin_amdgcn_wmma_f32_16x16x4_f32(false, a1, false, b[t], (short)0,
                                                        acc[1][t], false, false);
    }
  }

#pragma unroll
  for (int m = 0; m < 2; ++m) {
#pragma unroll
    for (int t = 0; t < 4; ++t) {
      const float bv = bias ? bias[col[t]] : 0.f;
      float* Cp = C + (size_t)(blockIdx.x * 32 + m * 16 + 8 * hi) * Nout + col[t];
#pragma unroll
      for (int r = 0; r < 8; ++r) {
        float v = acc[m][t][r] + bv;
        if (leaky) v = lrelu(v);
        Cp[(size_t)r * Nout] = v;
      }
    }
  }
}

// Scalar GEMM for odd shapes (K=6, K=2, Nout=6): one thread per output elem.
__global__ void gemm_small(const float* __restrict__ A, const float* __restrict__ W,
                           const float* __restrict__ bias, float* __restrict__ C,
                           int M, int K, int Nout, int leaky) {
  int t = blockIdx.x * blockDim.x + threadIdx.x;
  if (t >= M * Nout) return;
  int m = t / Nout, c = t - m * Nout;
  float s = bias ? bias[c] : 0.f;
  const float* Ar = A + (size_t)m * K;
  for (int k = 0; k < K; ++k) s += Ar[k] * W[(size_t)k * Nout + c];
  if (leaky) s = lrelu(s);
  C[t] = s;
}

// LayerNorm over rows of length D (<=256), one wave per row.
// Optional fused leaky after LN and fused residual add (out = resid + y).
__global__ void ln_kernel(const float* __restrict__ in, const float* __restrict__ g,
                          const float* __restrict__ b, const float* __restrict__ resid,
                          float* __restrict__ out, int D, int leaky) {
  const size_t rowD = (size_t)blockIdx.x * D;
  const int lane = threadIdx.x;
  float vals[8];
  float s = 0.f;
  int cnt = 0;
  for (int c = lane; c < D; c += 32) { float v = in[rowD + c]; vals[cnt++] = v; s += v; }
  s = wsum32(s);
  const float mu = s / D;
  float vs = 0.f;
  for (int i = 0; i < cnt; ++i) { float d = vals[i] - mu; vs += d * d; }
  vs = wsum32(vs);
  const float rstd = rsqrtf(vs / D + 1e-5f);
  cnt = 0;
  for (int c = lane; c < D; c += 32) {
    float y = (vals[cnt++] - mu) * rstd * g[c] + b[c];
    if (leaky) y = lrelu(y);
    if (resid) y += resid[rowD + c];
    out[rowD + c] = y;
  }
}

__global__ void fill_kernel(float* p, float v, int n) {
  int i = blockIdx.x * blockDim.x + threadIdx.x;
  if (i < n) p[i] = v;
}

// alpha[e,h] = dot(q[dst,h,:], k[src,h,:] + e[e,h,:]) / 8; atomic max over dst
// 8 lanes per (edge,head) unit.
__global__ void edge_alpha_kernel(const int* __restrict__ ei, const float* __restrict__ q,
                                  const float* __restrict__ k, const float* __restrict__ eE,
                                  float* __restrict__ alpha, float* __restrict__ amax, int E) {
  int t = blockIdx.x * blockDim.x + threadIdx.x;
  int unit = t >> 3, j = t & 7;
  if (unit >= E * kHeads) return;
  int e = unit >> 2, hh = unit & 3;
  int s = ei[e], d = ei[E + e];
  const float* qp = q + (size_t)d * kHC + hh * kHid;
  const float* kp = k + (size_t)s * kHC + hh * kHid;
  const float* ep = eE + (size_t)e * kHC + hh * kHid;
  float acc = 0.f;
  for (int c = j; c < kHid; c += 8) acc += qp[c] * (kp[c] + ep[c]);
  acc += __shfl_xor(acc, 1, 32);
  acc += __shfl_xor(acc, 2, 32);
  acc += __shfl_xor(acc, 4, 32);
  if (j == 0) {
    float a = acc * 0.125f;  // / sqrt(64)
    alpha[unit] = a;
    atomicMaxF(&amax[d * kHeads + hh], a);
  }
}

__global__ void edge_ea_kernel(const int* __restrict__ ei, const float* __restrict__ alpha,
                               const float* __restrict__ amax, float* __restrict__ ea,
                               float* __restrict__ den, int E) {
  int t = blockIdx.x * blockDim.x + threadIdx.x;
  if (t >= E * kHeads) return;
  int e = t >> 2, hh = t & 3;
  int d = ei[E + e];
  float v = expf(alpha[t] - amax[d * kHeads + hh]);
  ea[t] = v;
  atomicAdd(&den[d * kHeads + hh], v);
}

// msg = (v[src] + e) * attn, scatter-add into agg[dst]; one thread per element.
__global__ void edge_msg_kernel(const int* __restrict__ ei, const float* __restrict__ vv,
                                const float* __restrict__ eE, const float* __restrict__ ea,
                                const float* __restrict__ den, float* __restrict__ agg, int E) {
  int t = blockIdx.x * blockDim.x + threadIdx.x;
  if (t >= E * kHC) return;
  int e = t >> 8, r = t & 255;
  int hh = r >> 6;
  int s = ei[e], d = ei[E + e];
  float attn = ea[e * kHeads + hh] / (den[d * kHeads + hh] + 1e-16f);
  float val = (vv[(size_t)s * kHC + r] + eE[(size_t)e * kHC + r]) * attn;
  atomicAdd(&agg[(size_t)d * kHC + r], val);
}

// beta = sigmoid([out, xr, out-xr] @ Wbeta); hc = beta*xr + (1-beta)*out
__global__ void beta_gate_kernel(const float* __restrict__ agg, const float* __restrict__ xr,
                                 const float* __restrict__ Wb, float* __restrict__ hc) {
  const size_t base = (size_t)blockIdx.x * kHC;
  const int lane = threadIdx.x;
  float s = 0.f;
  for (int c = lane; c < kHC; c += 32) {
    float o = agg[base + c], r = xr[base + c];
    s += o * Wb[c] + r * Wb[kHC + c] + (o - r) * Wb[2 * kHC + c];
  }
  s = wsum32(s);
  float beta = 1.f / (1.f + expf(-s));
  for (int c = lane; c < kHC; c += 32) {
    float o = agg[base + c], r = xr[base + c];
    hc[base + c] = beta * r + (1.f - beta) * o;
  }
}

// I[dst] += Y * V[src]   (complex), V = Vm * exp(i*Va) from masked prediction
__global__ void engine_edge_kernel(const int* __restrict__ ei, const float* __restrict__ eat,
                                   const float* __restrict__ x, const int* __restrict__ mask,
                                   const float* __restrict__ tmp6, float* __restrict__ Ire,
                                   float* __restrict__ Iim, int E) {
  int e = blockIdx.x * blockDim.x + threadIdx.x;
  if (e >= E) return;
  int s = ei[e], d = ei[E + e];
  float Vm = mask[s * 6 + 4] ? x[s * 6 + 4] : tmp6[s * 6 + 4];
  float Va = mask[s * 6 + 5] ? x[s * 6 + 5] : tmp6[s * 6 + 5];
  float vr = Vm * cosf(Va), vi = Vm * sinf(Va);
  float yr = eat[e * 2], yi = eat[e * 2 + 1];
  atomicAdd(&Ire[d], yr * vr - yi * vi);
  atomicAdd(&Iim[d], yr * vi + yi * vr);
}

// res = V*conj(I) - ((Pg-Pd) + i(Qg-Qd))
__global__ void res_node_kernel(const float* __restrict__ x, const int* __restrict__ mask,
                                const float* __restrict__ tmp6, const float* __restrict__ Ire,
                                const float* __restrict__ Iim, float* __restrict__ resid, int Nn) {
  int n = blockIdx.x * blockDim.x + threadIdx.x;
  if (n >= Nn) return;
  float p[6];
#pragma unroll
  for (int i = 0; i < 6; ++i) p[i] = mask[n * 6 + i] ? x[n * 6 + i] : tmp6[n * 6 + i];
  float Pd = p[0], Qd = p[1], Pg = p[2], Qg = p[3], Vm = p[4], Va = p[5];
  float vr = Vm * cosf(Va), vi = Vm * sinf(Va);
  float ir = Ire[n], ii = Iim[n];
  float Sre = vr * ir + vi * ii;
  float Sim = vi * ir - vr * ii;
  resid[n * 2]     = Sre - (Pg - Pd);
  resid[n * 2 + 1] = Sim - (Qg - Qd);
}

// h[n,c] += leaky(resid[n] @ physW + physb)
__global__ void phys_add_kernel(const float* __restrict__ resid, const float* __restrict__ physW,
                                const float* __restrict__ physb, float* __restrict__ H, int Nn) {
  int t = blockIdx.x * blockDim.x + threadIdx.x;
  if (t >= Nn * kHC) return;
  int n = t >> 8, c = t & 255;
  float rr = resid[n * 2], ri = resid[n * 2 + 1];
  float v = rr * physW[c] + ri * physW[kHC + c] + physb[c];
  H[t] += lrelu(v);
}

// ---------------------------------------------------------------------------
// Launcher
// Input order: top-level setup_inputs dict insertion order; nested `params`
// pytree in jax tree_flatten (sorted-key DFS) order:
//   0 x, 1 pe, 2 edge_index, 3 edge_attr, 4 batch, 5 mask,
//   convs[i] (i=0..2): Wbeta, We, k.W, k.b, q.W, q.b, skip.W, skip.b, v.W, v.b
//   edge_proj: be, g, l1.W, l1.b, l2.W, l2.b
//   in_proj:   be, g, l1.W, l1.b, l2.W, l2.b
//   mlp:       be, g, l1.W, l1.b, l2.W, l2.b
//   norms[i]:  b, g
//   phys:      W, b
// ---------------------------------------------------------------------------
extern "C" void kernel_launch(void* const* d_in, const int* in_sizes, int n_in,
                              void* d_out, int out_size, void* d_ws, size_t ws_size,
                              hipStream_t stream) {
  (void)in_sizes; (void)n_in; (void)out_size; (void)ws_size;

  const float* x    = (const float*)d_in[0];
  const int*   ei   = (const int*)d_in[2];
  const float* eat  = (const float*)d_in[3];
  const int*   mask = (const int*)d_in[5];

  const float *cWbeta[3], *cWe[3], *ckW[3], *ckb[3], *cqW[3], *cqb[3],
              *csW[3], *csb[3], *cvW[3], *cvb[3];
  for (int i = 0; i < 3; ++i) {
    int base = 6 + i * 10;
    cWbeta[i] = (const float*)d_in[base + 0];
    cWe[i]    = (const float*)d_in[base + 1];
    ckW[i]    = (const float*)d_in[base + 2];
    ckb[i]    = (const float*)d_in[base + 3];
    cqW[i]    = (const float*)d_in[base + 4];
    cqb[i]    = (const float*)d_in[base + 5];
    csW[i]    = (const float*)d_in[base + 6];
    csb[i]    = (const float*)d_in[base + 7];
    cvW[i]    = (const float*)d_in[base + 8];
    cvb[i]    = (const float*)d_in[base + 9];
  }
  const float* ep_be  = (const float*)d_in[36];
  const float* ep_g   = (const float*)d_in[37];
  const float* ep_l1W = (const float*)d_in[38];
  const float* ep_l1b = (const float*)d_in[39];
  const float* ep_l2W = (const float*)d_in[40];
  const float* ep_l2b = (const float*)d_in[41];
  const float* ip_be  = (const float*)d_in[42];
  const float* ip_g   = (const float*)d_in[43];
  const float* ip_l1W = (const float*)d_in[44];
  const float* ip_l1b = (const float*)d_in[45];
  const float* ip_l2W = (const float*)d_in[46];
  const float* ip_l2b = (const float*)d_in[47];
  const float* m_be   = (const float*)d_in[48];
  const float* m_g    = (const float*)d_in[49];
  const float* m_l1W  = (const float*)d_in[50];
  const float* m_l1b  = (const float*)d_in[51];
  const float* m_l2W  = (const float*)d_in[52];
  const float* m_l2b  = (const float*)d_in[53];
  const float* n_b[3] = {(const float*)d_in[54], (const float*)d_in[56], (const float*)d_in[58]};
  const float* n_g[3] = {(const float*)d_in[55], (const float*)d_in[57], (const float*)d_in[59]};
  const float* physW  = (const float*)d_in[60];
  const float* physb  = (const float*)d_in[61];

  // workspace layout (floats)
  float* WSP = (float*)d_ws;
  size_t off = 0;
  auto alloc = [&](size_t n) { float* p = WSP + off; off += n; return p; };
  float* H     = alloc((size_t)kN * kHC);
  float* H64   = alloc((size_t)kN * kHid);
  float* EENC  = alloc((size_t)kE * kHid);
  float* EEm   = alloc((size_t)kE * kHC);
  float* Qb    = alloc((size_t)kN * kHC);
  float* Kb    = alloc((size_t)kN * kHC);
  float* Vb    = alloc((size_t)kN * kHC);
  float* XR    = alloc((size_t)kN * kHC);
  float* HCb   = alloc((size_t)kN * kHC);
  float* ALPHA = alloc((size_t)kE * kHeads);
  float* EA    = alloc((size_t)kE * kHeads);
  float* AMAX  = alloc((size_t)kN * kHeads);
  float* DEN   = alloc((size_t)kN * kHeads);
  float* AGG   = alloc((size_t)kN * kHC);
  float* Z64   = alloc((size_t)kE * kHid);   // proj / mlp intermediate
  float* TMP6  = alloc((size_t)kN * 6);
  float* IRE   = alloc((size_t)kN);
  float* IIM   = alloc((size_t)kN);
  float* RESID = alloc((size_t)kN * 2);

  // fast WMMA gemm: requires M%32==0, Nout%64==0, K in {64,256}
  auto gemm = [&](const float* A, const float* Wm, const float* bias, float* C,
                  int M, int K, int Nout, int leaky) {
    dim3 g(M / 32, Nout / 64);
    if (K == 64)
      gemm_f32_wmma<64><<<g, 32, 0, stream>>>(A, Wm, bias, C, Nout, leaky);
    else
      gemm_f32_wmma<256><<<g, 32, 0, stream>>>(A, Wm, bias, C, Nout, leaky);
  };
  auto gemmS = [&](const float* A, const float* Wm, const float* bias, float* C,
                   int M, int K, int Nout, int leaky) {
    int tot = M * Nout;
    gemm_small<<<(tot + 255) / 256, 256, 0, stream>>>(A, Wm, bias, C, M, K, Nout, leaky);
  };
  auto ln = [&](const float* in, const float* g, const float* b, const float* resid,
                float* out, int rows, int D, int leaky) {
    ln_kernel<<<rows, 32, 0, stream>>>(in, g, b, resid, out, D, leaky);
  };
  auto fill = [&](float* p, float v, int n) {
    fill_kernel<<<(n + 255) / 256, 256, 0, stream>>>(p, v, n);
  };

  // in_proj: x -> H64   (l1 has K=6 -> scalar kernel)
  gemmS(x, ip_l1W, ip_l1b, Z64, kN, 6, kHid, 1);
  gemm(Z64, ip_l2W, ip_l2b, H64, kN, kHid, kHid, 0);
  ln(H64, ip_g, ip_be, nullptr, H64, kN, kHid, 0);
  // edge_proj: edge_attr -> EENC   (l1 has K=2 -> scalar kernel)
  gemmS(eat, ep_l1W, ep_l1b, Z64, kE, 2, kHid, 1);
  gemm(Z64, ep_l2W, ep_l2b, EENC, kE, kHid, kHid, 0);
  ln(EENC, ep_g, ep_be, nullptr, EENC, kE, kHid, 0);

  for (int i = 0; i < 3; ++i) {
    const float* Ain = (i == 0) ? H64 : H;
    const int fin = (i == 0) ? kHid : kHC;
    gemm(Ain, cqW[i], cqb[i], Qb, kN, fin, kHC, 0);
    gemm(Ain, ckW[i], ckb[i], Kb, kN, fin, kHC, 0);
    gemm(Ain, cvW[i], cvb[i], Vb, kN, fin, kHC, 0);
    gemm(Ain, csW[i], csb[i], XR, kN, fin, kHC, 0);
    gemm(EENC, cWe[i], nullptr, EEm, kE, kHid, kHC, 0);

    fill(AMAX, -3.0e38f, kN * kHeads);
    fill(DEN, 0.f, kN * kHeads);
    fill(AGG, 0.f, kN * kHC);

    edge_alpha_kernel<<<(kE * kHeads * 8 + 255) / 256, 256, 0, stream>>>(
        ei, Qb, Kb, EEm, ALPHA, AMAX, kE);
    edge_ea_kernel<<<(kE * kHeads + 255) / 256, 256, 0, stream>>>(
        ei, ALPHA, AMAX, EA, DEN, kE);
    edge_msg_kernel<<<(kE * kHC + 255) / 256, 256, 0, stream>>>(
        ei, Vb, EEm, EA, DEN, AGG, kE);
    beta_gate_kernel<<<kN, 32, 0, stream>>>(AGG, XR, cWbeta[i], HCb);

    // h = (i ? h : 0) + leaky(LN(hc))
    ln(HCb, n_g[i], n_b[i], (i == 0) ? nullptr : H, H, kN, kHC, 1);

    // tmp = mlp_shared(h)
    gemm(H, m_l1W, m_l1b, Z64, kN, kHC, kHid, 0);
    ln(Z64, m_g, m_be, nullptr, Z64, kN, kHid, 1);
    gemmS(Z64, m_l2W, m_l2b, TMP6, kN, kHid, 6, 0);

    // physics residual
    fill(IRE, 0.f, kN);
    fill(IIM, 0.f, kN);
    engine_edge_kernel<<<(kE + 255) / 256, 256, 0, stream>>>(
        ei, eat, x, mask, TMP6, IRE, IIM, kE);
    res_node_kernel<<<(kN + 255) / 256, 256, 0, stream>>>(
        x, mask, TMP6, IRE, IIM, RESID, kN);
    phys_add_kernel<<<(kN * kHC + 255) / 256, 256, 0, stream>>>(
        RESID, physW, physb, H, kN);
  }

  // final mlp_shared -> d_out (N x 6, f32)
  gemm(H, m_l1W, m_l1b, Z64, kN, kHC, kHid, 0);
  ln(Z64, m_g, m_be, nullptr, Z64, kN, kHid, 1);
  gemmS(Z64, m_l2W, m_l2b, (float*)d_out, kN, kHid, 6, 0);
}